// SLDS_82918638617043
// MI455X (gfx1250) — compile-verified
//
#include <hip/hip_runtime.h>
#include <math.h>

static constexpr int K = 8;
static constexpr int H = 192;
static constexpr int D = 96;
static constexpr int TSTEPS = 16;
static constexpr float LOG_2PI = 1.8378770664093453f;

typedef __attribute__((ext_vector_type(2))) float v2f;
typedef __attribute__((ext_vector_type(8))) float v8f;

// ---------------- WMMA helpers (V_WMMA_F32_16X16X4_F32) ----------------
// A-frag 16x4: lanes 0-15 -> rows m0..m0+15 with K={k,k+1}; lanes 16-31 same rows K={k+2,k+3}
// B-frag 4x16: lane%16 -> column, VGPR0/1 -> K pair selected by lane/16
// C/D frag 16x16 f32: v8f; row = m0 + r + 8*(lane>>4), col = n0 + (lane&15)

// 32x32 macro-tile: 2x2 of 16x16 accumulators; per k-step 4 fragment loads feed 4 WMMAs.
__device__ __forceinline__ void wmma_tile32(const float* __restrict__ Amat, int lda,
                                            bool transA,
                                            const float* __restrict__ Bmat, int ldb,
                                            int m0, int n0, int kdim, int lane,
                                            v8f acc[2][2]) {
  const int half = lane >> 4;
  const int lo = lane & 15;
  for (int k = 0; k < kdim; k += 4) {
    const int kb = k + (half << 1);
    v2f a0, a1, b0, b1;
    if (!transA) {
      const float* ap = Amat + (size_t)(m0 + lo) * lda + kb;
      a0.x = ap[0]; a0.y = ap[1];
      const float* ap2 = ap + (size_t)16 * lda;
      a1.x = ap2[0]; a1.y = ap2[1];
    } else {
      // operand is stored-matrix transposed: op[m][k] = Amat[k][m]
      const float* ap = Amat + (size_t)kb * lda + (m0 + lo);
      a0.x = ap[0];  a0.y = ap[lda];
      a1.x = ap[16]; a1.y = ap[lda + 16];
    }
    const float* bp = Bmat + (size_t)kb * ldb + (n0 + lo);
    b0.x = bp[0];  b0.y = bp[ldb];
    b1.x = bp[16]; b1.y = bp[ldb + 16];
    acc[0][0] = __builtin_amdgcn_wmma_f32_16x16x4_f32(false, a0, false, b0, (short)0,
                                                      acc[0][0], false, false);
    acc[0][1] = __builtin_amdgcn_wmma_f32_16x16x4_f32(false, a0, false, b1, (short)0,
                                                      acc[0][1], false, false);
    acc[1][0] = __builtin_amdgcn_wmma_f32_16x16x4_f32(false, a1, false, b0, (short)0,
                                                      acc[1][0], false, false);
    acc[1][1] = __builtin_amdgcn_wmma_f32_16x16x4_f32(false, a1, false, b1, (short)0,
                                                      acc[1][1], false, false);
  }
}

__device__ __forceinline__ void wmma_store_tile(float* __restrict__ Cmat, int ldc,
                                                int m0, int n0, int lane, v8f c) {
  const int col = n0 + (lane & 15);
  const int rb = m0 + ((lane >> 4) << 3);
#pragma unroll
  for (int r = 0; r < 8; r++) Cmat[(size_t)(rb + r) * ldc + col] = c[r];
}

__device__ __forceinline__ void wmma_store32(float* __restrict__ Cmat, int ldc,
                                             int m0, int n0, int lane, v8f acc[2][2]) {
#pragma unroll
  for (int mi = 0; mi < 2; mi++)
#pragma unroll
    for (int ni = 0; ni < 2; ni++)
      wmma_store_tile(Cmat, ldc, m0 + 16 * mi, n0 + 16 * ni, lane, acc[mi][ni]);
}

// ---------------- small setup kernels ----------------
__global__ void k_logA(const float* __restrict__ logits, float* __restrict__ logA) {
  __shared__ float rl[K];
  int tid = threadIdx.x;
  if (tid < K) {
    float mx = -1e30f;
    for (int j = 0; j < K; j++) mx = fmaxf(mx, logits[tid * K + j]);
    float s = 0.f;
    for (int j = 0; j < K; j++) s += expf(logits[tid * K + j] - mx);
    rl[tid] = mx + logf(s);
  }
  __syncthreads();
  if (tid < K * K) logA[tid] = logits[tid] - rl[tid / K];
}

__global__ void k_init(const float* __restrict__ logA, float* __restrict__ Pbar,
                       float* __restrict__ mbar, float* __restrict__ w_pred) {
  int j = blockIdx.x, tid = threadIdx.x;
  float* Pb = Pbar + (size_t)j * H * H;
  for (int idx = tid; idx < H * H; idx += blockDim.x)
    Pb[idx] = ((idx / H) == (idx % H)) ? 1.f : 0.f;
  for (int h = tid; h < H; h += blockDim.x) mbar[j * H + h] = 0.f;
  if (tid == 0) w_pred[j] = logA[j];  // logA[0][j]
}

// mu[i][j][g] = sum_h m[i][h] * A[j][h][g]
__global__ void k_mu(const float* __restrict__ m, const float* __restrict__ A,
                     float* __restrict__ mu) {
  int b = blockIdx.x;
  int i = b / K, j = b % K;
  int g = threadIdx.x;
  const float* mi = m + i * H;
  const float* Aj = A + (size_t)j * H * H;
  float s = 0.f;
  for (int h = 0; h < H; h++) s += mi[h] * Aj[(size_t)h * H + g];
  mu[(size_t)b * H + g] = s;
}

// l = w[i] + logA[i][j]; w_pred[j] = lse_i l; p[i][j] = exp(l - w_pred[j])
__global__ void k_weights(const float* __restrict__ w, const float* __restrict__ logA,
                          float* __restrict__ w_pred, float* __restrict__ p) {
  __shared__ float ls[K * K];
  __shared__ float wp[K];
  int tid = threadIdx.x;
  int i = tid / K, j = tid % K;
  float l = w[i] + logA[i * K + j];
  ls[tid] = l;
  __syncthreads();
  if (i == 0) {
    float mx = -1e30f;
    for (int ii = 0; ii < K; ii++) mx = fmaxf(mx, ls[ii * K + j]);
    float s = 0.f;
    for (int ii = 0; ii < K; ii++) s += expf(ls[ii * K + j] - mx);
    wp[j] = mx + logf(s);
    w_pred[j] = wp[j];
  }
  __syncthreads();
  p[tid] = expf(l - wp[j]);
}

__global__ void k_mbar(const float* __restrict__ p, const float* __restrict__ mu,
                       float* __restrict__ mbar) {
  int j = blockIdx.x, g = threadIdx.x;
  float s = 0.f;
  for (int i = 0; i < K; i++) s += p[i * K + j] * mu[(size_t)(i * K + j) * H + g];
  mbar[j * H + g] = s;
}

// ---------------- predict + moment match (WMMA heavy) ----------------
// Per j (one WG, 8 waves): Pbar[j] = diag(exp(q)) + sum_i p_ij*(A_j^T P_i A_j + dm dm^T)
__global__ void k_predict(const float* __restrict__ A, const float* __restrict__ logQ,
                          const float* __restrict__ p, const float* __restrict__ mu,
                          const float* __restrict__ mbar, float* __restrict__ PAws,
                          float* __restrict__ Pbar, const float* __restrict__ Pst) {
  int j = blockIdx.x, tid = threadIdx.x;
  int lane = tid & 31, wave = tid >> 5;
  const float* Aj = A + (size_t)j * H * H;
  float* PA = PAws + (size_t)j * H * H;
  float* Pb = Pbar + (size_t)j * H * H;
  const float* mbj = mbar + j * H;
  for (int idx = tid; idx < H * H; idx += blockDim.x)
    Pb[idx] = ((idx / H) == (idx % H)) ? expf(logQ[idx / H]) : 0.f;
  __threadfence();
  __syncthreads();
  constexpr int MT = (H / 32) * (H / 32);  // 36 macro-tiles of 32x32
  for (int i = 0; i < K; i++) {
    const float* Pi = Pst + (size_t)i * H * H;
    const float pij = p[i * K + j];
    const float* mui = mu + (size_t)(i * K + j) * H;
    // phase 1: PA = P_i @ A_j   (192x192x192)
    for (int t = wave; t < MT; t += 8) {
      int mo = (t / (H / 32)) * 32, no = (t % (H / 32)) * 32;
      v8f acc[2][2] = {};
      wmma_tile32(Pi, H, false, Aj, H, mo, no, H, lane, acc);
      wmma_store32(PA, H, mo, no, lane, acc);
    }
    __threadfence();
    __syncthreads();
    // phase 2: Pbar += p_ij * (A_j^T @ PA + dm dm^T)
    for (int t = wave; t < MT; t += 8) {
      int mo = (t / (H / 32)) * 32, no = (t % (H / 32)) * 32;
      v8f acc[2][2] = {};
      wmma_tile32(Aj, H, true, PA, H, mo, no, H, lane, acc);
#pragma unroll
      for (int mi = 0; mi < 2; mi++)
#pragma unroll
        for (int ni = 0; ni < 2; ni++) {
          int col = no + 16 * ni + (lane & 15);
          int rb = mo + 16 * mi + ((lane >> 4) << 3);
          float dml = mui[col] - mbj[col];
#pragma unroll
          for (int r = 0; r < 8; r++) {
            int row = rb + r;
            float dmg = mui[row] - mbj[row];
            Pb[(size_t)row * H + col] += pij * (acc[mi][ni][r] + dmg * dml);
          }
        }
    }
    __threadfence();
    __syncthreads();
  }
}

// ---------------- observation update ----------------
__global__ void k_PC(const float* __restrict__ Pbar, const float* __restrict__ C,
                     float* __restrict__ PC) {
  int j = blockIdx.x, tid = threadIdx.x;
  int lane = tid & 31, wave = tid >> 5;
  const float* Pb = Pbar + (size_t)j * H * H;
  float* PCj = PC + (size_t)j * H * D;
  for (int t = wave; t < (H / 32) * (D / 32); t += 8) {
    int mo = (t / (D / 32)) * 32, no = (t % (D / 32)) * 32;
    v8f acc[2][2] = {};
    wmma_tile32(Pb, H, false, C, D, mo, no, H, lane, acc);
    wmma_store32(PCj, D, mo, no, lane, acc);
  }
}

__global__ void k_S(const float* __restrict__ C, const float* __restrict__ PC,
                    const float* __restrict__ logR, float* __restrict__ S) {
  int j = blockIdx.x, tid = threadIdx.x;
  int lane = tid & 31, wave = tid >> 5;
  const float* PCj = PC + (size_t)j * H * D;
  float* Sj = S + (size_t)j * D * D;
  for (int t = wave; t < (D / 32) * (D / 32); t += 8) {
    int mo = (t / (D / 32)) * 32, no = (t % (D / 32)) * 32;
    v8f acc[2][2] = {};
    wmma_tile32(C, D, true, PCj, D, mo, no, H, lane, acc);  // C^T @ PC
#pragma unroll
    for (int mi = 0; mi < 2; mi++)
#pragma unroll
      for (int ni = 0; ni < 2; ni++) {
        int col = no + 16 * ni + (lane & 15);
        int rb = mo + 16 * mi + ((lane >> 4) << 3);
#pragma unroll
        for (int r = 0; r < 8; r++) {
          int row = rb + r;
          float v = acc[mi][ni][r];
          if (row == col) v += expf(logR[row]);
          Sj[(size_t)row * D + col] = v;
        }
      }
  }
}

// Cholesky of S (LDS), innovation, alpha solve, logZ, w update, m_post; L written back
__global__ void k_obs_finish(float* __restrict__ S, const float* __restrict__ mbar,
                             const float* __restrict__ C, const float* __restrict__ PC,
                             const float* __restrict__ y, const float* __restrict__ w_pred,
                             float* __restrict__ w, float* __restrict__ m) {
  __shared__ float Ls[D * D];
  __shared__ float vv[D], zz[D], xx[D];
  int j = blockIdx.x, tid = threadIdx.x;
  float* Sg = S + (size_t)j * D * D;
  const float* mb = mbar + j * H;
  const float* PCj = PC + (size_t)j * H * D;
  for (int idx = tid; idx < D * D; idx += blockDim.x) Ls[idx] = Sg[idx];
  if (tid < D) {
    float s = y[tid];
    for (int h = 0; h < H; h++) s -= mb[h] * C[(size_t)h * D + tid];
    vv[tid] = s;
  }
  __syncthreads();
  // lower Cholesky in LDS
  for (int k = 0; k < D; k++) {
    if (tid == 0) Ls[k * D + k] = sqrtf(Ls[k * D + k]);
    __syncthreads();
    float lkk = Ls[k * D + k];
    for (int i = k + 1 + tid; i < D; i += blockDim.x) Ls[i * D + k] /= lkk;
    __syncthreads();
    int rem = D - k - 1;
    for (int idx = tid; idx < rem * rem; idx += blockDim.x) {
      int r = idx / rem, c = idx % rem;
      int ii = k + 1 + r, jc = k + 1 + c;
      if (jc <= ii) Ls[ii * D + jc] -= Ls[ii * D + k] * Ls[jc * D + k];
    }
    __syncthreads();
  }
  // forward solve L z = v
  if (tid < D) zz[tid] = vv[tid];
  __syncthreads();
  for (int k = 0; k < D; k++) {
    if (tid == 0) zz[k] /= Ls[k * D + k];
    __syncthreads();
    float zk = zz[k];
    for (int i = k + 1 + tid; i < D; i += blockDim.x) zz[i] -= Ls[i * D + k] * zk;
    __syncthreads();
  }
  // back solve L^T x = z
  if (tid < D) xx[tid] = zz[tid];
  __syncthreads();
  for (int k = D - 1; k >= 0; k--) {
    if (tid == 0) xx[k] /= Ls[k * D + k];
    __syncthreads();
    float xk = xx[k];
    for (int i = tid; i < k; i += blockDim.x) xx[i] -= Ls[k * D + i] * xk;
    __syncthreads();
  }
  if (tid == 0) {
    float sl = 0.f, va = 0.f;
    for (int d2 = 0; d2 < D; d2++) {
      sl += logf(Ls[d2 * D + d2]);
      va += vv[d2] * xx[d2];
    }
    w[j] = w_pred[j] + (-0.5f * ((float)D * LOG_2PI + 2.f * sl + va));
  }
  __syncthreads();
  if (tid < H) {
    float s = mb[tid];
    for (int d2 = 0; d2 < D; d2++) s += PCj[(size_t)tid * D + d2] * xx[d2];
    m[j * H + tid] = s;
  }
  for (int idx = tid; idx < D * D; idx += blockDim.x) Sg[idx] = Ls[idx];
}

// X = S^{-1} PC^T  (one RHS column of PC^T per thread)
__global__ void k_X(const float* __restrict__ S, const float* __restrict__ PC,
                    float* __restrict__ X) {
  __shared__ float Ls[D * D];
  int j = blockIdx.x, tid = threadIdx.x;
  const float* Sg = S + (size_t)j * D * D;
  const float* PCj = PC + (size_t)j * H * D;
  float* Xj = X + (size_t)j * D * H;
  for (int idx = tid; idx < D * D; idx += blockDim.x) Ls[idx] = Sg[idx];
  __syncthreads();
  if (tid < H) {
    float yv[D];
    for (int i = 0; i < D; i++) {
      float s = PCj[(size_t)tid * D + i];
      for (int k2 = 0; k2 < i; k2++) s -= Ls[i * D + k2] * yv[k2];
      yv[i] = s / Ls[i * D + i];
    }
    for (int i = D - 1; i >= 0; i--) {
      float s = yv[i];
      for (int k2 = i + 1; k2 < D; k2++) s -= Ls[k2 * D + i] * yv[k2];
      yv[i] = s / Ls[i * D + i];
    }
    for (int i = 0; i < D; i++) Xj[(size_t)i * H + tid] = yv[i];
  }
}

// Praw = Pbar - PC @ X
__global__ void k_Ppost(const float* __restrict__ Pbar, const float* __restrict__ PC,
                        const float* __restrict__ X, float* __restrict__ Praw) {
  int j = blockIdx.x, tid = threadIdx.x;
  int lane = tid & 31, wave = tid >> 5;
  const float* Pb = Pbar + (size_t)j * H * H;
  const float* PCj = PC + (size_t)j * H * D;
  const float* Xj = X + (size_t)j * D * H;
  float* Pr = Praw + (size_t)j * H * H;
  for (int t = wave; t < (H / 32) * (H / 32); t += 8) {
    int mo = (t / (H / 32)) * 32, no = (t % (H / 32)) * 32;
    v8f acc[2][2] = {};
    wmma_tile32(PCj, D, false, Xj, H, mo, no, D, lane, acc);
#pragma unroll
    for (int mi = 0; mi < 2; mi++)
#pragma unroll
      for (int ni = 0; ni < 2; ni++) {
        int col = no + 16 * ni + (lane & 15);
        int rb = mo + 16 * mi + ((lane >> 4) << 3);
#pragma unroll
        for (int r = 0; r < 8; r++) {
          int row = rb + r;
          Pr[(size_t)row * H + col] = Pb[(size_t)row * H + col] - acc[mi][ni][r];
        }
      }
  }
}

__global__ void k_symm(const float* __restrict__ Praw, float* __restrict__ P) {
  int j = blockIdx.x, tid = threadIdx.x;
  const float* Pr = Praw + (size_t)j * H * H;
  float* Pj = P + (size_t)j * H * H;
  for (int idx = tid; idx < H * H; idx += blockDim.x) {
    int g = idx / H, l = idx % H;
    Pj[idx] = 0.5f * (Pr[idx] + Pr[(size_t)l * H + g]);
  }
}

__global__ void k_final(const float* __restrict__ w, float* __restrict__ out) {
  if (threadIdx.x == 0) {
    float mx = -1e30f;
    for (int i = 0; i < K; i++) mx = fmaxf(mx, w[i]);
    float s = 0.f;
    for (int i = 0; i < K; i++) s += expf(w[i] - mx);
    out[0] = mx + logf(s);
  }
}

extern "C" void kernel_launch(void* const* d_in, const int* in_sizes, int n_in,
                              void* d_out, int out_size, void* d_ws, size_t ws_size,
                              hipStream_t stream) {
  (void)in_sizes; (void)n_in; (void)out_size; (void)ws_size;
  const float* data   = (const float*)d_in[0];  // (T, D)
  const float* logits = (const float*)d_in[1];  // (K, K)
  const float* A      = (const float*)d_in[2];  // (K, H, H)
  const float* logQ   = (const float*)d_in[3];  // (H,)
  const float* C      = (const float*)d_in[4];  // (H, D)
  const float* logR   = (const float*)d_in[5];  // (D,)
  float* out = (float*)d_out;
  float* ws = (float*)d_ws;

  size_t off = 0;
  float* logA   = ws + off; off += K * K;
  float* w      = ws + off; off += K;
  float* w_pred = ws + off; off += K;
  float* p      = ws + off; off += K * K;
  float* m      = ws + off; off += (size_t)K * H;
  float* mu     = ws + off; off += (size_t)K * K * H;
  float* mbar   = ws + off; off += (size_t)K * H;
  float* Pst    = ws + off; off += (size_t)K * H * H;
  float* Pbar   = ws + off; off += (size_t)K * H * H;
  float* PAws   = ws + off; off += (size_t)K * H * H;
  float* Praw   = ws + off; off += (size_t)K * H * H;
  float* PC     = ws + off; off += (size_t)K * H * D;
  float* S      = ws + off; off += (size_t)K * D * D;
  float* X      = ws + off; off += (size_t)K * D * H;

  k_logA<<<1, 64, 0, stream>>>(logits, logA);
  // t = 0: P = I, m = 0, prior = logA[0]
  k_init<<<K, 256, 0, stream>>>(logA, Pbar, mbar, w_pred);
  k_PC<<<K, 256, 0, stream>>>(Pbar, C, PC);
  k_S<<<K, 256, 0, stream>>>(C, PC, logR, S);
  k_obs_finish<<<K, 192, 0, stream>>>(S, mbar, C, PC, data, w_pred, w, m);
  k_X<<<K, 192, 0, stream>>>(S, PC, X);
  k_Ppost<<<K, 256, 0, stream>>>(Pbar, PC, X, Praw);
  k_symm<<<K, 256, 0, stream>>>(Praw, Pst);

  for (int t = 1; t < TSTEPS; t++) {
    const float* y = data + (size_t)t * D;
    k_mu<<<K * K, H, 0, stream>>>(m, A, mu);
    k_weights<<<1, K * K, 0, stream>>>(w, logA, w_pred, p);
    k_mbar<<<K, H, 0, stream>>>(p, mu, mbar);
    k_predict<<<K, 256, 0, stream>>>(A, logQ, p, mu, mbar, PAws, Pbar, Pst);
    k_PC<<<K, 256, 0, stream>>>(Pbar, C, PC);
    k_S<<<K, 256, 0, stream>>>(C, PC, logR, S);
    k_obs_finish<<<K, 192, 0, stream>>>(S, mbar, C, PC, y, w_pred, w, m);
    k_X<<<K, 192, 0, stream>>>(S, PC, X);
    k_Ppost<<<K, 256, 0, stream>>>(Pbar, PC, X, Praw);
    k_symm<<<K, 256, 0, stream>>>(Praw, Pst);
  }
  k_final<<<1, 32, 0, stream>>>(w, out);
}